// SOM_7052336300201
// MI455X (gfx1250) — compile-verified
//
#include <hip/hip_runtime.h>

#define BATCH_N 8192
#define KNODES  1024
#define DDIM    12288
#define EPSV    1e-6f

typedef __attribute__((ext_vector_type(2))) float v2f;
typedef __attribute__((ext_vector_type(8))) float v8f;

// ---------------------------------------------------------------- init
__global__ void som_init_min(unsigned long long* dmin) {
  int i = blockIdx.x * blockDim.x + threadIdx.x;
  if (i < BATCH_N) dmin[i] = ~0ull;
}

// ----------------------------------------------- per-node constant term
// const[k] = ||w_k||^2 - 2*eps*sum(w_k)
__global__ void som_node_const(const float* __restrict__ w, float* __restrict__ cst) {
  __shared__ float ssq[256];
  __shared__ float ssm[256];
  const int k = blockIdx.x;
  const float* row = w + (size_t)k * DDIM;
  float sq = 0.f, sm = 0.f;
  for (int d = threadIdx.x; d < DDIM; d += 256) {
    float v = row[d];
    sq += v * v;
    sm += v;
  }
  ssq[threadIdx.x] = sq;
  ssm[threadIdx.x] = sm;
  __syncthreads();
  for (int s = 128; s > 0; s >>= 1) {
    if (threadIdx.x < s) {
      ssq[threadIdx.x] += ssq[threadIdx.x + s];
      ssm[threadIdx.x] += ssm[threadIdx.x + s];
    }
    __syncthreads();
  }
  if (threadIdx.x == 0) cst[k] = ssq[0] - 2.0f * EPSV * ssm[0];
}

// ------------------------------------------------- weights transpose
// wT[d*1024 + k] = w[k*DDIM + d]   (makes B-fragment loads coalesced)
__global__ void som_transpose(const float* __restrict__ w, float* __restrict__ wT) {
  __shared__ float tile[32][33];
  const int kb = blockIdx.x * 32;
  const int db = blockIdx.y * 32;
  const int tx = threadIdx.x & 31;
  const int ty = threadIdx.x >> 5;   // 0..7
  #pragma unroll
  for (int j = 0; j < 32; j += 8)
    tile[ty + j][tx] = w[(size_t)(kb + ty + j) * DDIM + db + tx];
  __syncthreads();
  #pragma unroll
  for (int j = 0; j < 32; j += 8)
    wT[(size_t)(db + ty + j) * KNODES + kb + tx] = tile[tx][ty + j];
}

// order-preserving float -> uint, concat with index: atomicMin == argmin
__device__ __forceinline__ unsigned long long sort_key(float s, unsigned idx) {
  unsigned u = __float_as_uint(s);
  u = (u & 0x80000000u) ? ~u : (u | 0x80000000u);
  return ((unsigned long long)u << 32) | (unsigned long long)idx;
}

// ------------------------------------------------- fused GEMM + argmin
// Workgroup: 64 rows x 1024 cols. 8 waves; wave owns 128 cols, 4 row tiles.
// f32 WMMA 16x16x4: A = 2 VGPRs (lane->M, {vgpr,half}->K), C = 8 VGPRs.
template <bool TRANSPOSED>
__global__ __launch_bounds__(256, 1)
void som_gemm_argmin(const float* __restrict__ x, const float* __restrict__ wmat,
                     const float* __restrict__ cst,
                     unsigned long long* __restrict__ dmin) {
  __shared__ float lds_a[64 * 34];           // 64 rows, padded stride 34

  const int row0 = blockIdx.x * 64;
  const int t    = threadIdx.x;
  const int wave = t >> 5;
  const int lane = t & 31;
  const int half = lane >> 4;                // which 16-lane half
  const int lr   = lane & 15;
  const int n0   = wave * 128;               // this wave's column base

  v8f acc[4][8];
  {
    v8f z;
    #pragma unroll
    for (int i = 0; i < 8; i++) z[i] = 0.f;
    #pragma unroll
    for (int rt = 0; rt < 4; rt++)
      #pragma unroll
      for (int ct = 0; ct < 8; ct++) acc[rt][ct] = z;
  }

  const int lrow = t >> 2;                   // 0..63  (coop A load)
  const int lcol = (t & 3) * 8;              // 0,8,16,24

  for (int kk = 0; kk < DDIM; kk += 32) {
    __syncthreads();
    const float* xs = x + (size_t)(row0 + lrow) * DDIM + kk + lcol;
    float4 p0 = *(const float4*)(xs);
    float4 p1 = *(const float4*)(xs + 4);
    float* d = &lds_a[lrow * 34 + lcol];
    d[0] = p0.x; d[1] = p0.y; d[2] = p0.z; d[3] = p0.w;
    d[4] = p1.x; d[5] = p1.y; d[6] = p1.z; d[7] = p1.w;
    __syncthreads();

    #pragma unroll
    for (int ks = 0; ks < 32; ks += 4) {
      v2f a[4], b[8];
      #pragma unroll
      for (int rt = 0; rt < 4; rt++)
        a[rt] = *(const v2f*)&lds_a[(rt * 16 + lr) * 34 + ks + 2 * half];

      const int kd = kk + ks + 2 * half;
      #pragma unroll
      for (int ct = 0; ct < 8; ct++) {
        const int col = n0 + ct * 16 + lr;
        if (TRANSPOSED) {
          b[ct][0] = wmat[(size_t)kd * KNODES + col];
          b[ct][1] = wmat[(size_t)(kd + 1) * KNODES + col];
        } else {
          b[ct] = *(const v2f*)(wmat + (size_t)col * DDIM + kd);
        }
      }

      #pragma unroll
      for (int rt = 0; rt < 4; rt++)
        #pragma unroll
        for (int ct = 0; ct < 8; ct++)
          acc[rt][ct] = __builtin_amdgcn_wmma_f32_16x16x4_f32(
              false, a[rt], false, b[ct], (short)0, acc[rt][ct], false, false);
    }
  }

  // ---- epilogue: score = const[col] - 2*cross; argmin over 1024 cols
  float cv[8];
  #pragma unroll
  for (int ct = 0; ct < 8; ct++) cv[ct] = cst[n0 + ct * 16 + lr];

  #pragma unroll
  for (int rt = 0; rt < 4; rt++) {
    #pragma unroll
    for (int v = 0; v < 8; v++) {
      float best = 3.4e38f;
      int   bidx = 0;
      #pragma unroll
      for (int ct = 0; ct < 8; ct++) {
        float s = cv[ct] - 2.0f * acc[rt][ct][v];
        int idx = n0 + ct * 16 + lr;
        if (s < best || (s == best && idx < bidx)) { best = s; bidx = idx; }
      }
      // reduce across the 16 lanes of this half (rows differ per half)
      #pragma unroll
      for (int m = 1; m < 16; m <<= 1) {
        float os = __shfl_xor(best, m, 32);
        int   oi = __shfl_xor(bidx, m, 32);
        if (os < best || (os == best && oi < bidx)) { best = os; bidx = oi; }
      }
      if (lr == 0) {
        int row = row0 + rt * 16 + v + 8 * half;
        atomicMin(&dmin[row], sort_key(best, (unsigned)bidx));
      }
    }
  }
}

// ---------------------------------------------------------------- gather
// out[b] = weights[k%32, k/32]  ->  node = (k%32)*32 + k/32
__global__ void som_gather(const float* __restrict__ w,
                           const unsigned long long* __restrict__ dmin,
                           float* __restrict__ out) {
  const int b = blockIdx.x;
  const unsigned k = (unsigned)(dmin[b] & 0xFFFFFFFFull);
  const int node = (int)((k & 31u) * 32u + (k >> 5));
  const float4* src = (const float4*)(w + (size_t)node * DDIM);
  float4* dst = (float4*)(out + (size_t)b * DDIM);
  for (int i = threadIdx.x; i < DDIM / 4; i += blockDim.x) dst[i] = src[i];
}

// ---------------------------------------------------------------- launch
extern "C" void kernel_launch(void* const* d_in, const int* in_sizes, int n_in,
                              void* d_out, int out_size, void* d_ws, size_t ws_size,
                              hipStream_t stream) {
  const float* x = (const float*)d_in[0];
  const float* w = (const float*)d_in[1];
  float* out = (float*)d_out;

  char* ws = (char*)d_ws;
  unsigned long long* dmin = (unsigned long long*)ws;           // 64 KB
  float* cst = (float*)(ws + 65536);                            // 4 KB
  float* wT  = (float*)(ws + 65536 + 16384);                    // 48 MB (optional)
  const size_t need_t = 65536 + 16384 + (size_t)DDIM * KNODES * sizeof(float);
  const bool use_t = ws_size >= need_t;

  som_init_min<<<(BATCH_N + 255) / 256, 256, 0, stream>>>(dmin);
  som_node_const<<<KNODES, 256, 0, stream>>>(w, cst);

  if (use_t) {
    dim3 g(KNODES / 32, DDIM / 32);
    som_transpose<<<g, 256, 0, stream>>>(w, wT);
    som_gemm_argmin<true><<<BATCH_N / 64, 256, 0, stream>>>(x, wT, cst, dmin);
  } else {
    som_gemm_argmin<false><<<BATCH_N / 64, 256, 0, stream>>>(x, w, cst, dmin);
  }

  som_gather<<<BATCH_N, 256, 0, stream>>>(w, dmin, out);
}